// CapsuleLayer_47364899340443
// MI455X (gfx1250) — compile-verified
//
#include <hip/hip_runtime.h>
#include <math.h>

// ---------------------------------------------------------------------------
// Capsule dynamic routing for MI455X (gfx1250, wave32, WMMA).
//   B=64 batches, I=2048 input caps, P=16, J=32 output caps, D=16, 3 routings.
// u_hat (256 MB if materialized) is never written to memory: every routing
// pass recomputes its 16x16 tiles in registers with V_WMMA_F32_16X16X4_F32
// from L2-resident operands (W = 64 MB, x^T = 8 MB; global L2 = 192 MB).
// Routing logits are linear in u_hat (b starts at 0), so each pass only needs
// vcomb = sum of previously squashed outputs (zeros -> uniform softmax in r0).
//
// Operand layouts: every WMMA element pair is one global_load_b64:
//   A (16x4, M=batch):  x^T[i][b][p]   -> lane m reads pair (p=k0, k0+1)
//   B (4x16, N=d):      W[i][j][d][p]  -> lane n reads pair (p=k0, k0+1)
// Logits use a reduce-scatter butterfly (lane ln ends holding L[b=ln>>1]);
// softmax over j is done once per b-row by waves 0..15 with full-wave
// butterflies instead of 32x-redundant serial LDS loops.
// ---------------------------------------------------------------------------

#define B_BATCH 64
#define I_CAPS  2048
#define P_DIM   16
#define J_CAPS  32
#define D_DIM   16
#define IPB     16            // i's per block in the routing pass
#define LROW    33            // padded LDS row stride (bank-conflict-free)
#define EPS_F   1e-7f

typedef float v2f __attribute__((ext_vector_type(2)));
typedef float v8f __attribute__((ext_vector_type(8)));

__device__ inline void atomic_add_f32(float* p, float v) {
    unsafeAtomicAdd(p, v);    // global_atomic_add_f32 on gfx1250
}

// ---- x^T: xt[i][b][p] = inputs[b][i][p]  (p stays innermost) --------------
__global__ void caps_transpose_x(const float* __restrict__ in, float* __restrict__ xt) {
    int t = blockIdx.x * blockDim.x + threadIdx.x;     // over I*B*P
    if (t >= I_CAPS * B_BATCH * P_DIM) return;
    int i = t / (B_BATCH * P_DIM);
    int r = t % (B_BATCH * P_DIM);
    int b = r / P_DIM;
    int p = r % P_DIM;
    xt[t] = in[(b * I_CAPS + i) * P_DIM + p];
}

__global__ void caps_zero(float* __restrict__ p, int n) {
    int t = blockIdx.x * blockDim.x + threadIdx.x;
    if (t < n) p[t] = 0.0f;
}

// ---------------------------------------------------------------------------
// One routing pass:
//   per (b,i): L[j] = dot(u_hat[b,i,j,:], vcomb[b,j,:]); c = softmax_j(L);
//   s[b,j,d] += c[j] * u_hat[b,i,j,d]
// Block: 32 waves (one per output capsule j); blockIdx.x = batch tile of 16;
// blockIdx.y = i-chunk.  u_hat tiles live only in registers.
// ---------------------------------------------------------------------------
__global__ __launch_bounds__(1024)
void caps_route_pass(const float* __restrict__ xt,     // [I][B][P]
                     const float* __restrict__ w,      // [I][J][D][P] (original)
                     const float* __restrict__ vcomb,  // [B][J][D]
                     float* __restrict__ s_out)        // [B][J][D]
{
    __shared__ float lg2[16 * LROW];                   // [b_local][j] logits -> c

    const int lane = threadIdx.x & 31;
    const int j    = threadIdx.x >> 5;                 // wave id == output capsule
    const int b0   = blockIdx.x * 16;                  // batch tile base
    const int i0   = blockIdx.y * IPB;
    const int half = (lane >= 16) ? 1 : 0;             // WMMA lane half
    const int ln   = lane & 15;                        // n-index (d) / m-index

    // vcomb tile in C/D layout: vv[v] <-> (b = b0 + v + 8*half, d = ln)
    float vv[8];
#pragma unroll
    for (int v = 0; v < 8; ++v)
        vv[v] = vcomb[((b0 + v + 8 * half) * J_CAPS + j) * D_DIM + ln];

    v8f s_acc = {0.f, 0.f, 0.f, 0.f, 0.f, 0.f, 0.f, 0.f};

    for (int ii = 0; ii < IPB; ++ii) {
        const int i = i0 + ii;
        const float* xa = xt + (i * B_BATCH + b0 + ln) * P_DIM;          // row of x^T
        const float* wb = w + ((i * J_CAPS + j) * D_DIM + ln) * P_DIM;   // row of W[i][j]
        if (i + 1 < I_CAPS)  // prefetch next i's 1KB W tile (global_prefetch_b8)
            __builtin_prefetch(w + (((i + 1) * J_CAPS + j) * D_DIM) * P_DIM, 0, 3);

        // u_hat tile [16 b x 16 d] for this (i, j): 4 chained K=4 fp32 WMMAs
        v8f acc = {0.f, 0.f, 0.f, 0.f, 0.f, 0.f, 0.f, 0.f};
#pragma unroll
        for (int ko = 0; ko < 4; ++ko) {
            const int k0 = 4 * ko + 2 * half;  // lanes 0-15: K={k,k+1}; 16-31: K={k+2,k+3}
            const v2f a = *(const v2f*)(xa + k0);   // one global_load_b64
            const v2f b = *(const v2f*)(wb + k0);   // one global_load_b64
            acc = __builtin_amdgcn_wmma_f32_16x16x4_f32(
                false, a, false, b, (short)0, acc, false, false);
        }

        // ---- logits L[b] = sum_d u[b,d]*vcomb[b,d]: reduce-scatter butterfly.
        // After the stages, lane ln holds L[b0 + 8*half + (ln>>1)].
        float xv[8];
#pragma unroll
        for (int v = 0; v < 8; ++v) xv[v] = acc[v] * vv[v];

        float y[4];
#pragma unroll
        for (int v = 0; v < 4; ++v) {           // stage mask=8: merge v / v+4
            float s = (lane & 8) ? xv[v + 4] : xv[v];
            float t = (lane & 8) ? xv[v]     : xv[v + 4];
            y[v] = s + __shfl_xor(t, 8, 32);
        }
        float z[2];
#pragma unroll
        for (int v = 0; v < 2; ++v) {           // stage mask=4: merge v / v+2
            float s = (lane & 4) ? y[v + 2] : y[v];
            float t = (lane & 4) ? y[v]     : y[v + 2];
            z[v] = s + __shfl_xor(t, 4, 32);
        }
        float s3 = (lane & 2) ? z[1] : z[0];    // stage mask=2: merge 0 / 1
        float t3 = (lane & 2) ? z[0] : z[1];
        float u = s3 + __shfl_xor(t3, 2, 32);
        u += __shfl_xor(u, 1, 32);              // stage mask=1: finish the sum

        if ((lane & 1) == 0)                    // one ds_store_b32, 16 rows
            lg2[(8 * half + (ln >> 1)) * LROW + j] = u;
        __syncthreads();

        // ---- softmax over j, one wave per b-row (waves 0..15; lane = j)
        if (j < 16) {
            const int row = j * LROW;
            const float lgv = lg2[row + lane];
            float mm = lgv;
#pragma unroll
            for (int mask = 1; mask < 32; mask <<= 1)
                mm = fmaxf(mm, __shfl_xor(mm, mask, 32));
            const float e = __expf(lgv - mm);
            float Zs = e;
#pragma unroll
            for (int mask = 1; mask < 32; mask <<= 1)
                Zs += __shfl_xor(Zs, mask, 32);
            lg2[row + lane] = e / Zs;           // in-place: logits -> c[b][j]
        }
        __syncthreads();

        // ---- s_acc[b,d] += c[b][j] * u[b,d]  (broadcast ds_loads)
#pragma unroll
        for (int v = 0; v < 8; ++v) {
            const float cv = lg2[(v + 8 * half) * LROW + j];
            s_acc[v] += cv * acc[v];
        }
    }

    // partial s for this (batch tile, j, i-chunk) -> global fp32 atomics
#pragma unroll
    for (int v = 0; v < 8; ++v)
        atomic_add_f32(&s_out[((b0 + v + 8 * half) * J_CAPS + j) * D_DIM + ln],
                       s_acc[v]);
}

// ---- squash: v = (n2/(1+n2)) * s / sqrt(n2);  vcomb += v; maybe emit out --
__global__ void caps_squash(const float* __restrict__ s, float* __restrict__ vcomb,
                            float* __restrict__ out, int write_out) {
    int t = blockIdx.x * blockDim.x + threadIdx.x;     // over B*J
    if (t >= B_BATCH * J_CAPS) return;
    const float* sp = s + t * D_DIM;
    float vals[D_DIM];
    float n2 = EPS_F;
#pragma unroll
    for (int d = 0; d < D_DIM; ++d) { vals[d] = sp[d]; n2 += vals[d] * vals[d]; }
    const float scale = n2 / ((1.0f + n2) * sqrtf(n2));
#pragma unroll
    for (int d = 0; d < D_DIM; ++d) {
        const float v = scale * vals[d];
        vcomb[t * D_DIM + d] += v;
        if (write_out) out[t * D_DIM + d] = v;
    }
}

extern "C" void kernel_launch(void* const* d_in, const int* in_sizes, int n_in,
                              void* d_out, int out_size, void* d_ws, size_t ws_size,
                              hipStream_t stream) {
    const float* inputs = (const float*)d_in[0];   // [64,2048,16] f32
    const float* W      = (const float*)d_in[1];   // [2048,32,16,16] f32
    float* out = (float*)d_out;                    // [64,32,16] f32

    float* ws    = (float*)d_ws;
    float* xt    = ws;                                     // 2,097,152 f (8 MB)
    float* sbuf  = xt + I_CAPS * B_BATCH * P_DIM;          //    32,768 f
    float* vcomb = sbuf + B_BATCH * J_CAPS * D_DIM;        //    32,768 f

    const int NX = I_CAPS * B_BATCH * P_DIM;
    const int NS = B_BATCH * J_CAPS * D_DIM;

    caps_transpose_x<<<(NX + 255) / 256, 256, 0, stream>>>(inputs, xt);
    caps_zero<<<(NS + 255) / 256, 256, 0, stream>>>(vcomb, NS);

    const dim3 grid(B_BATCH / 16, I_CAPS / IPB);           // 4 x 128 blocks
    for (int r = 0; r < 3; ++r) {
        caps_zero<<<(NS + 255) / 256, 256, 0, stream>>>(sbuf, NS);
        caps_route_pass<<<grid, 1024, 0, stream>>>(xt, W, vcomb, sbuf);
        caps_squash<<<(B_BATCH * J_CAPS + 255) / 256, 256, 0, stream>>>(
            sbuf, vcomb, out, (r == 2) ? 1 : 0);
    }
}